// EMAVectorQuantizer_74423193305765
// MI455X (gfx1250) — compile-verified
//
#include <hip/hip_runtime.h>
#include <math.h>

typedef __attribute__((ext_vector_type(2))) float v2f;
typedef __attribute__((ext_vector_type(4))) float v4f;
typedef __attribute__((ext_vector_type(8))) float v8f;

#define N_TOK   65536
#define N_E     4096
#define E_DIM   64
#define SPAT    16384           /* 16*32*32 spatial per batch */
#define BSTRIDE (64*16384)      /* batch stride in z (c-major) */
#define BETA    0.25f
#define DECAY   0.99f
#define EPSV    1e-5f

#define CHUNK   256             /* codes staged in LDS per step */
#define ROWP    68              /* padded LDS row stride (floats) */

/* d_out layout (floats), reference tuple return order */
#define OUT_OFF_OUT   0
#define OUT_OFF_LOSS  4194304
#define OUT_OFF_PERP  4194305
#define OUT_OFF_IDX   4194306
#define OUT_OFF_NEMB  4259842
#define OUT_OFF_NCS   4521986
#define OUT_OFF_NEA   4526082

/* workspace layout (floats) */
#define WS_LOSS   0
#define WS_NSUM   1
#define WS_PLOG   2
#define WS_ENORM  16
#define WS_COUNTS (WS_ENORM + N_E)
#define WS_ESUM   (WS_COUNTS + N_E)

/* ---------------- K0: zero accumulators, compute code norms ------------- */
__global__ void k0_init(const float* __restrict__ embed_w, float* __restrict__ ws) {
  int i = blockIdx.x * blockDim.x + threadIdx.x;     /* 262144 threads */
  if (i < 16) ws[i] = 0.0f;
  if (i < N_E) {
    ws[WS_COUNTS + i] = 0.0f;
    const float* e = embed_w + i * E_DIM;
    float s = 0.0f;
#pragma unroll 8
    for (int c = 0; c < E_DIM; ++c) { float v = e[c]; s += v * v; }
    ws[WS_ENORM + i] = s;
  }
  if (i < N_E * E_DIM) ws[WS_ESUM + i] = 0.0f;
}

/* ---------------- K1: fp32 WMMA distance GEMM + fused argmin ------------ */
/* 8 waves/WG share LDS-staged codebook chunks; one wave = 16 tokens.      */
/* score(t,n) = ||e_n||^2 - 2 * <z_t, e_n>  (||z_t||^2 constant in n)      */
__global__ void __launch_bounds__(256)
k1_argmin(const float* __restrict__ z, const float* __restrict__ embed_w,
          const float* __restrict__ ws, float* __restrict__ out_idx) {
  __shared__ float sB[CHUNK * ROWP];
  __shared__ float sEn[CHUNK];

  const int tid   = threadIdx.x;
  const int lane  = tid & 31;
  const int tbase = (blockIdx.x * 8 + (tid >> 5)) * 16;
  const int lo = lane & 15;
  const int hi = lane >> 4;

  /* A tile: token tbase+lo, dims [hi*32, hi*32+32). zf[t][c] = z[b][c][s] */
  float az[32];
  {
    const int t = tbase + lo;
    const float* zb = z + (size_t)(t >> 14) * BSTRIDE + (t & (SPAT - 1));
    const int cbase = hi * 32;
#pragma unroll
    for (int i = 0; i < 32; ++i) az[i] = zb[(size_t)(cbase + i) * SPAT];
  }

  float mins[8];
  int   midx[8];
#pragma unroll
  for (int r = 0; r < 8; ++r) { mins[r] = 3.4e38f; midx[r] = 0; }

  for (int chunk = 0; chunk < N_E / CHUNK; ++chunk) {
    __syncthreads();
    /* cooperative stage: thread tid copies one full code row + its norm */
    {
      const float* src = embed_w + (size_t)(chunk * CHUNK + tid) * E_DIM;
      float* dst = sB + tid * ROWP;
#pragma unroll
      for (int i = 0; i < 16; ++i)
        *(v4f*)(dst + 4 * i) = *(const v4f*)(src + 4 * i);
      sEn[tid] = ws[WS_ENORM + chunk * CHUNK + tid];
    }
    __syncthreads();

    for (int t16 = 0; t16 < CHUNK / 16; ++t16) {
      const int nl = t16 * 16 + lo;
      /* B tile from LDS: code nl, dims [hi*32, hi*32+32) */
      const float* brow = sB + nl * ROWP + hi * 32;
      v4f bv[8];
#pragma unroll
      for (int i = 0; i < 8; ++i) bv[i] = *(const v4f*)(brow + 4 * i);

      v8f acc = {0.f, 0.f, 0.f, 0.f, 0.f, 0.f, 0.f, 0.f};
#pragma unroll
      for (int g = 0; g < 16; ++g) {
        v2f a = { az[2 * g], az[2 * g + 1] };
        v2f b = { bv[g >> 1][(2 * g) & 3], bv[g >> 1][(2 * g + 1) & 3] };
        acc = __builtin_amdgcn_wmma_f32_16x16x4_f32(
            /*neg_a=*/false, a, /*neg_b=*/false, b,
            /*c_mod=*/(short)0, acc, /*reuse_a=*/false, /*reuse_b=*/false);
      }
      const float en = sEn[nl];
      const int   n  = chunk * CHUNK + nl;
#pragma unroll
      for (int r = 0; r < 8; ++r) {
        float s = en - 2.0f * acc[r];
        if (s < mins[r]) { mins[r] = s; midx[r] = n; }
      }
    }
  }

  /* reduce over the 16 N-lanes of each half (tokens live in VGPR slots) */
#pragma unroll
  for (int r = 0; r < 8; ++r) {
    float v = mins[r]; int ix = midx[r];
#pragma unroll
    for (int off = 1; off < 16; off <<= 1) {
      float ov = __shfl_xor(v, off, 32);
      int   oi = __shfl_xor(ix, off, 32);
      if (ov < v || (ov == v && oi < ix)) { v = ov; ix = oi; }
    }
    mins[r] = v; midx[r] = ix;
  }
  if (lo == 0) {
#pragma unroll
    for (int r = 0; r < 8; ++r) {
      int t = tbase + r + 8 * hi;   /* D-layout: M = r + 8*hi */
      out_idx[t] = (float)midx[r];
    }
  }
}

/* ------- K2: gather z_q, write out, loss partials, counts, embed_sum ---- */
__global__ void k2_gather(const float* __restrict__ z,
                          const float* __restrict__ embed_w,
                          const float* __restrict__ idxf,
                          float* __restrict__ out, float* __restrict__ ws) {
  const int t = blockIdx.x * blockDim.x + threadIdx.x;   /* 65536 tokens */
  const int code = (int)idxf[t];
  const float* e  = embed_w + (size_t)code * E_DIM;
  const float* zb = z   + (size_t)(t >> 14) * BSTRIDE + (t & (SPAT - 1));
  float*       ob = out + (size_t)(t >> 14) * BSTRIDE + (t & (SPAT - 1));
  float* esum = ws + WS_ESUM + (size_t)code * E_DIM;
  float lsum = 0.0f;
#pragma unroll 8
  for (int c = 0; c < E_DIM; ++c) {
    float zv = zb[(size_t)c * SPAT];
    float ev = e[c];
    float d  = ev - zv;
    lsum += d * d;
    ob[(size_t)c * SPAT] = ev;                 /* straight-through forward */
    atomicAdd(&esum[c], zv);                   /* one_hot.T @ zf           */
  }
  atomicAdd(&ws[WS_COUNTS + code], 1.0f);
#pragma unroll
  for (int off = 16; off >= 1; off >>= 1) lsum += __shfl_xor(lsum, off, 32);
  if ((threadIdx.x & 31) == 0) atomicAdd(&ws[WS_LOSS], lsum);
}

/* ---------------- K3: per-code new_cs, n-sum, perplexity partial -------- */
__global__ void k3_percode(const float* __restrict__ cluster_size,
                           float* __restrict__ out_ncs, float* __restrict__ ws) {
  const int i = blockIdx.x * blockDim.x + threadIdx.x;   /* 4096 codes */
  float c   = ws[WS_COUNTS + i];
  float ncs = cluster_size[i] * DECAY + (1.0f - DECAY) * c;
  out_ncs[i] = ncs;
  float p  = c / (float)N_TOK;
  float pl = p * logf(p + 1e-10f);
  float ns = ncs, ps = pl;
#pragma unroll
  for (int off = 16; off >= 1; off >>= 1) {
    ns += __shfl_xor(ns, off, 32);
    ps += __shfl_xor(ps, off, 32);
  }
  if ((threadIdx.x & 31) == 0) {
    atomicAdd(&ws[WS_NSUM], ns);
    atomicAdd(&ws[WS_PLOG], ps);
  }
}

/* ---------------- K5: new_ea / new_embed + scalar outputs --------------- */
__global__ void k5_final(const float* __restrict__ embed_avg,
                         const float* __restrict__ ws,
                         const float* __restrict__ ncs,
                         float* __restrict__ d_out) {
  const int i = blockIdx.x * blockDim.x + threadIdx.x;   /* 262144 */
  float es = ws[WS_ESUM + i];
  float ea = embed_avg[i] * DECAY + (1.0f - DECAY) * es;
  d_out[OUT_OFF_NEA + i] = ea;
  float n = ws[WS_NSUM];
  float c = ncs[i >> 6];
  float smoothed = (c + EPSV) / (n + (float)N_E * EPSV) * n;
  d_out[OUT_OFF_NEMB + i] = ea / smoothed;
  if (i == 0) {
    d_out[OUT_OFF_LOSS] = BETA * ws[WS_LOSS] / (float)(N_TOK * E_DIM);
    d_out[OUT_OFF_PERP] = expf(-ws[WS_PLOG]);
  }
}

extern "C" void kernel_launch(void* const* d_in, const int* in_sizes, int n_in,
                              void* d_out, int out_size, void* d_ws, size_t ws_size,
                              hipStream_t stream) {
  (void)in_sizes; (void)n_in; (void)out_size; (void)ws_size;
  const float* z            = (const float*)d_in[0];
  const float* embed_w      = (const float*)d_in[1];
  const float* cluster_size = (const float*)d_in[2];
  const float* embed_avg    = (const float*)d_in[3];
  float* out = (float*)d_out;
  float* ws  = (float*)d_ws;

  k0_init  <<<(N_E * E_DIM + 255) / 256, 256, 0, stream>>>(embed_w, ws);
  k1_argmin<<<N_TOK / 16 / 8, 256, 0, stream>>>(z, embed_w, ws, out + OUT_OFF_IDX);
  k2_gather<<<N_TOK / 256, 256, 0, stream>>>(z, embed_w, out + OUT_OFF_IDX, out, ws);
  k3_percode<<<N_E / 256, 256, 0, stream>>>(cluster_size, out + OUT_OFF_NCS, ws);
  k5_final <<<(N_E * E_DIM + 255) / 256, 256, 0, stream>>>(embed_avg, ws, out + OUT_OFF_NCS, out);
}